// DisparityEstimation_8658654068880
// MI455X (gfx1250) — compile-verified
//
#include <hip/hip_runtime.h>
#include <hip/hip_bf16.h>

// CDNA5 / gfx1250 disparity softmax-regression.
//
// Roofline: 402.7MB input @ 23.3TB/s => ~17.3us floor. Naive per-element VALU
// (exp + 3 weighted-sum FMAs) exceeds that, so the three reductions
// (sum e, sum e*d', sum e*d'^2) are done on the matrix pipe with
// V_WMMA_F32_16X16X4_F32 (full f32, no precision loss), leaving VALU only the
// exp2 (which co-executes with XDL WMMA). d' is centered at +/-95.5 to avoid
// E[d^2]-E[d]^2 cancellation.

typedef __attribute__((ext_vector_type(2))) float v2f;
typedef __attribute__((ext_vector_type(8))) float v8f;

#define B_DIM 4
#define D_DIM 192
#define H_DIM 256
#define W_DIM 512
#define HW    (H_DIM * W_DIM)        // 131072
#define DHW   (D_DIM * HW)
#define BHW   (B_DIM * HW)           // 524288

__global__ __launch_bounds__(256, 1)
void disp_softmax_wmma_kernel(const float* __restrict__ in, float* __restrict__ out)
{
    const int wg   = blockIdx.x;          // b*H + h  (0..1023) -> one W-row
    const int tid  = threadIdx.x;
    const int wave = tid >> 5;            // 8 waves, 64 pixels each
    const int lane = tid & 31;
    const int m16  = lane & 15;           // A row / B column (pixel-in-group)
    const int khalf = lane >> 4;          // K half: lanes 16-31 hold K=2,3

    const int b  = wg >> 8;               // wg / 256
    const int h  = wg & 255;
    const int w0 = wave * 64;             // wave's first pixel in the row
    // per-lane base pointer at (b, d = khalf*2, h, w0 + m16)
    const float* p = in + (size_t)b * DHW + (size_t)h * W_DIM + (w0 + m16)
                        + (size_t)(khalf * 2) * HW;

    // A-matrix row selectors: row0 -> 1, row1 -> d', row2 -> d'^2, rows 3..15 -> 0
    const float f0 = (m16 == 0) ? 1.0f : 0.0f;
    const float f1 = (m16 == 1) ? 1.0f : 0.0f;
    const float f2 = (m16 == 2) ? 1.0f : 0.0f;

    // centered disparity for this lane's two K slots (K = 2*khalf + vgpr)
    float dk0 = -95.5f + (float)(khalf * 2);
    float dk1 = dk0 + 1.0f;

    v8f c0 = {}, c1 = {}, c2 = {}, c3 = {};   // accumulators, 4 pixel groups

    const float LOG2E = 1.4426950408889634f;
    const float CLMP  = 101.0f;   // exp arg ~70 in base-2 domain: no overflow,
                                  // exact vs max-subtracted softmax for sane inputs

    for (int cIdx = 0; cIdx < 48; ++cIdx) {   // 48 chunks x K=4 = D=192
        // Rebuild A (16x4) from current centered d: a = f0 + f1*d + f2*d^2
        v2f a;
        a[0] = fmaf(f2, dk0 * dk0, fmaf(f1, dk0, f0));
        a[1] = fmaf(f2, dk1 * dk1, fmaf(f1, dk1, f0));

        // Load + exp for 4 pixel groups (two coalesced 64B segments per load)
        v2f bg0, bg1, bg2, bg3;
        {
            float x00 = p[0 * 16],      x01 = p[0 * 16 + HW];
            float x10 = p[1 * 16],      x11 = p[1 * 16 + HW];
            float x20 = p[2 * 16],      x21 = p[2 * 16 + HW];
            float x30 = p[3 * 16],      x31 = p[3 * 16 + HW];
            bg0[0] = __builtin_amdgcn_exp2f(fminf(x00 * LOG2E, CLMP));
            bg0[1] = __builtin_amdgcn_exp2f(fminf(x01 * LOG2E, CLMP));
            bg1[0] = __builtin_amdgcn_exp2f(fminf(x10 * LOG2E, CLMP));
            bg1[1] = __builtin_amdgcn_exp2f(fminf(x11 * LOG2E, CLMP));
            bg2[0] = __builtin_amdgcn_exp2f(fminf(x20 * LOG2E, CLMP));
            bg2[1] = __builtin_amdgcn_exp2f(fminf(x21 * LOG2E, CLMP));
            bg3[0] = __builtin_amdgcn_exp2f(fminf(x30 * LOG2E, CLMP));
            bg3[1] = __builtin_amdgcn_exp2f(fminf(x31 * LOG2E, CLMP));
        }

        // One shared A, four B tiles -> four f32 WMMAs
        c0 = __builtin_amdgcn_wmma_f32_16x16x4_f32(false, a, false, bg0, (short)0, c0, false, false);
        c1 = __builtin_amdgcn_wmma_f32_16x16x4_f32(false, a, false, bg1, (short)0, c1, false, false);
        c2 = __builtin_amdgcn_wmma_f32_16x16x4_f32(false, a, false, bg2, (short)0, c2, false, false);
        c3 = __builtin_amdgcn_wmma_f32_16x16x4_f32(false, a, false, bg3, (short)0, c3, false, false);

        dk0 += 4.0f;
        dk1 += 4.0f;
        p   += 4 * HW;
    }

    // Epilogue: lanes 0-15 hold C rows 0..7; rows 0,1,2 = S0, S1, S2 per pixel.
    if (lane < 16) {
        const int pixBase = wg * W_DIM + w0 + m16;
        const v8f cc[4] = {c0, c1, c2, c3};
        #pragma unroll
        for (int g = 0; g < 4; ++g) {
            float s0 = cc[g][0];
            float s1 = cc[g][1];
            float s2 = cc[g][2];
            float inv  = __builtin_amdgcn_rcpf(s0);       // v_rcp_f32
            float mc   = s1 * inv;                        // centered mean
            float var  = fmaf(-mc, mc, s2 * inv);         // E[d'^2] - E[d']^2
            var = fmaxf(var, 0.0f);
            // log(x) = log2(x) * ln2 via v_log_f32
            float lv = __builtin_amdgcn_logf(var + 1e-6f) * 0.6931471805599453f;
            int pix = pixBase + g * 16;
            out[pix]       = mc + 95.5f;                  // un-center mean
            out[BHW + pix] = lv;
        }
    }
}

extern "C" void kernel_launch(void* const* d_in, const int* in_sizes, int n_in,
                              void* d_out, int out_size, void* d_ws, size_t ws_size,
                              hipStream_t stream) {
    (void)in_sizes; (void)n_in; (void)d_ws; (void)ws_size; (void)out_size;
    const float* cost = (const float*)d_in[0];
    float* out = (float*)d_out;
    // one workgroup per (b, h) row: 4*256 = 1024 blocks of 256 threads (8 waves)
    disp_softmax_wmma_kernel<<<dim3(B_DIM * H_DIM), dim3(256), 0, stream>>>(cost, out);
}